// GCNEncoder_72541997629772
// MI455X (gfx1250) — compile-verified
//
#include <hip/hip_runtime.h>

typedef float v2f __attribute__((ext_vector_type(2)));
typedef float v8f __attribute__((ext_vector_type(8)));

#define DH 128

// ---------------- utility kernels ----------------

__global__ void k_fill(float* __restrict__ p, float v, int n) {
  int i = blockIdx.x * blockDim.x + threadIdx.x;
  if (i < n) p[i] = v;
}

__global__ void k_deg_accum(const int* __restrict__ dst, float* __restrict__ deg, int E) {
  int i = blockIdx.x * blockDim.x + threadIdx.x;
  if (i < E) atomicAdd(&deg[dst[i]], 1.0f);
}

__global__ void k_rsqrt_inplace(float* __restrict__ p, int n) {
  int i = blockIdx.x * blockDim.x + threadIdx.x;
  if (i < n) p[i] = rsqrtf(p[i]);
}

// ---------------- fp32 WMMA GEMM: C[M x Nc] = A[M x K] @ B[K x Nc] ----------------
// One wave32 computes one 16x16 output tile with V_WMMA_F32_16X16X4_F32.
// A frag (16x4): lane 0-15 hold M=lane, K={k,k+1}; lanes 16-31 hold M=lane-16, K={k+2,k+3}.
// B frag (4x16): lane 0-15 hold N=lane, K={k,k+1}; lanes 16-31 hold N=lane-16, K={k+2,k+3}.
// C/D (16x16):  VGPR v, lanes 0-15 -> M=v, lanes 16-31 -> M=v+8; N = lane%16.
__global__ void k_gemm_wmma_f32(const float* __restrict__ A, const float* __restrict__ B,
                                float* __restrict__ C, int M, int K, int Nc) {
  const int gtid  = blockIdx.x * blockDim.x + threadIdx.x;
  const int wave  = gtid >> 5;
  const int lane  = threadIdx.x & 31;
  const int tilesN = Nc >> 4;
  const int tileM  = wave / tilesN;
  const int tileN  = wave - tileM * tilesN;
  if ((tileM << 4) >= M) return;            // wave-uniform guard (EXEC stays all-1s)

  const int row = (tileM << 4) + (lane & 15);
  const int col = (tileN << 4) + (lane & 15);
  const int kh  = (lane >> 4) << 1;         // 0 for lanes 0-15, 2 for lanes 16-31

  const float* __restrict__ ap = A + (size_t)row * K + kh;
  const float* __restrict__ bp = B + (size_t)kh * Nc + col;

  v8f acc = {};
  for (int k = 0; k < K; k += 4) {
    v2f a, b;
    a.x = ap[k];
    a.y = ap[k + 1];
    b.x = bp[(size_t)k * Nc];
    b.y = bp[(size_t)(k + 1) * Nc];
    acc = __builtin_amdgcn_wmma_f32_16x16x4_f32(
        /*neg_a=*/false, a, /*neg_b=*/false, b,
        /*c_mod=*/(short)0, acc, /*reuse_a=*/false, /*reuse_b=*/false);
  }

  const int rbase = (tileM << 4) + ((lane >> 4) << 3);
  float* __restrict__ cp = C + (size_t)rbase * Nc + col;
#pragma unroll
  for (int v = 0; v < 8; ++v) cp[(size_t)v * Nc] = acc[v];
}

// ---------------- GCN conv: self-loop init + edge scatter ----------------

// out[i] = m[i] * dinv[row]^2   (self-loop term, also initializes out)
__global__ void k_selfloop_init(const float* __restrict__ m, const float* __restrict__ dinv,
                                float* __restrict__ out, int N, int D) {
  int i = blockIdx.x * blockDim.x + threadIdx.x;
  if (i < N * D) {
    float di = dinv[i / D];
    out[i] = m[i] * di * di;
  }
}

// D/4 lanes per edge (one wave when D=128); each lane handles a float4 chunk of the feature row
__global__ void k_edge_scatter(const float* __restrict__ m, const int* __restrict__ src,
                               const int* __restrict__ dst, const float* __restrict__ dinv,
                               float* __restrict__ out, int E, int D) {
  const int chunks = D >> 2;  // float4 chunks per row (32 for D=128, 16 for D=64)
  long long idx = (long long)blockIdx.x * blockDim.x + threadIdx.x;
  if (idx >= (long long)E * chunks) return;
  const int e = (int)(idx / chunks);
  const int c = ((int)(idx - (long long)e * chunks)) << 2;
  const int s = src[e];
  const int d = dst[e];
  const float coef = dinv[s] * dinv[d];
  const float4 v = *(const float4*)(m + (size_t)s * D + c);
  float* __restrict__ op = out + (size_t)d * D + c;
  atomicAdd(op + 0, v.x * coef);
  atomicAdd(op + 1, v.y * coef);
  atomicAdd(op + 2, v.z * coef);
  atomicAdd(op + 3, v.w * coef);
}

// ---------------- BatchNorm (+ReLU) ----------------

// blockDim.x == 128; thread t owns column t; coalesced row-major reads
__global__ void k_bn_partial(const float* __restrict__ h, float* __restrict__ sum,
                             float* __restrict__ sumsq, int N, int rowsPerBlock) {
  const int col = threadIdx.x;
  int r0 = blockIdx.x * rowsPerBlock;
  int r1 = r0 + rowsPerBlock;
  if (r1 > N) r1 = N;
  float s = 0.f, s2 = 0.f;
  for (int r = r0; r < r1; ++r) {
    float v = h[(size_t)r * DH + col];
    s += v;
    s2 += v * v;
  }
  atomicAdd(&sum[col], s);
  atomicAdd(&sumsq[col], s2);
}

__global__ void k_bn_finalize(const float* __restrict__ sum, const float* __restrict__ sumsq,
                              const float* __restrict__ gamma, const float* __restrict__ beta,
                              float* __restrict__ scale, float* __restrict__ shift, int N) {
  const int c = threadIdx.x;
  const float inv_n = 1.0f / (float)N;
  const float mean = sum[c] * inv_n;
  const float var = sumsq[c] * inv_n - mean * mean;
  const float rstd = rsqrtf(var + 1e-5f);
  const float sc = rstd * gamma[c];
  scale[c] = sc;
  shift[c] = beta[c] - mean * sc;
}

__global__ void k_bn_apply_relu(float* __restrict__ h, const float* __restrict__ scale,
                                const float* __restrict__ shift, int total) {
  int i = blockIdx.x * blockDim.x + threadIdx.x;
  if (i < total) {
    const int c = i & (DH - 1);
    float v = h[i] * scale[c] + shift[c];
    h[i] = v > 0.f ? v : 0.f;
  }
}

// ---------------- host-side orchestration ----------------

static inline int cdiv(long long a, int b) { return (int)((a + b - 1) / b); }

static void run_gemm(const float* A, const float* W, float* C, int N, int K, int Nc,
                     hipStream_t stream) {
  const long long tiles = (long long)(N / 16) * (Nc / 16);
  const long long threads = tiles * 32;
  k_gemm_wmma_f32<<<cdiv(threads, 256), 256, 0, stream>>>(A, W, C, N, K, Nc);
}

static void run_conv(const float* m, const int* src, const int* dst, const float* dinv,
                     float* out, int N, int E, int D, hipStream_t stream) {
  k_selfloop_init<<<cdiv((long long)N * D, 256), 256, 0, stream>>>(m, dinv, out, N, D);
  const long long sthreads = (long long)E * (D >> 2);
  k_edge_scatter<<<cdiv(sthreads, 256), 256, 0, stream>>>(m, src, dst, dinv, out, E, D);
}

static void run_bn_relu(float* h, const float* gamma, const float* beta, float* stats, int N,
                        hipStream_t stream) {
  float* sum = stats;
  float* sumsq = stats + 128;
  float* scale = stats + 256;
  float* shift = stats + 384;
  k_fill<<<1, 256, 0, stream>>>(sum, 0.0f, 256);  // zero sum+sumsq (contiguous)
  const int rowsPerBlock = 256;
  k_bn_partial<<<cdiv(N, rowsPerBlock), 128, 0, stream>>>(h, sum, sumsq, N, rowsPerBlock);
  k_bn_finalize<<<1, 128, 0, stream>>>(sum, sumsq, gamma, beta, scale, shift, N);
  k_bn_apply_relu<<<cdiv((long long)N * DH, 256), 256, 0, stream>>>(h, scale, shift, N * DH);
}

extern "C" void kernel_launch(void* const* d_in, const int* in_sizes, int n_in,
                              void* d_out, int out_size, void* d_ws, size_t ws_size,
                              hipStream_t stream) {
  const float* x      = (const float*)d_in[0];
  const int*   eidx   = (const int*)d_in[1];
  const float* W0     = (const float*)d_in[2];
  const float* W1     = (const float*)d_in[3];
  const float* Wmu    = (const float*)d_in[4];
  const float* Wlv    = (const float*)d_in[5];
  const float* gamma0 = (const float*)d_in[6];
  const float* beta0  = (const float*)d_in[7];
  const float* gamma1 = (const float*)d_in[8];
  const float* beta1  = (const float*)d_in[9];

  const int N = in_sizes[0] / DH;     // 50000
  const int E = in_sizes[1] / 2;      // 800000
  const int DLAT = in_sizes[4] / DH;  // W_mu is [128, D_LAT] -> 64
  const int* src = eidx;
  const int* dst = eidx + E;

  float* ws    = (float*)d_ws;
  float* dinv  = ws;                       // [N]
  float* stats = ws + N;                   // [512]: sum, sumsq, scale, shift
  float* m     = stats + 512;              // [N*128]
  float* h     = m + (size_t)N * DH;       // [N*128]
  float* zmean = (float*)d_out;            // [N*DLAT]
  float* zlog  = zmean + (size_t)N * DLAT; // [N*DLAT]

  // degrees with self-loop: deg = 1 + segment_sum(1, dst); dinv = rsqrt(deg)
  k_fill<<<cdiv(N, 256), 256, 0, stream>>>(dinv, 1.0f, N);
  k_deg_accum<<<cdiv(E, 256), 256, 0, stream>>>(dst, dinv, E);
  k_rsqrt_inplace<<<cdiv(N, 256), 256, 0, stream>>>(dinv, N);

  // layer 0: h = relu(bn(conv(x @ W0)))
  run_gemm(x, W0, m, N, DH, DH, stream);
  run_conv(m, src, dst, dinv, h, N, E, DH, stream);
  run_bn_relu(h, gamma0, beta0, stats, N, stream);

  // layer 1: h = relu(bn(conv(h @ W1)))   (h free after gemm; conv writes back into h)
  run_gemm(h, W1, m, N, DH, DH, stream);
  run_conv(m, src, dst, dinv, h, N, E, DH, stream);
  run_bn_relu(h, gamma1, beta1, stats, N, stream);

  // latent heads -> d_out
  run_gemm(h, Wmu, m, N, DH, DLAT, stream);
  run_conv(m, src, dst, dinv, zmean, N, E, DLAT, stream);

  run_gemm(h, Wlv, m, N, DH, DLAT, stream);
  run_conv(m, src, dst, dinv, zlog, N, E, DLAT, stream);
}